// TaskAwareMOERouter_8143257993600
// MI455X (gfx1250) — compile-verified
//
#include <hip/hip_runtime.h>
#include <hip/hip_bf16.h>
#include <math.h>

// ---------------------------------------------------------------------------
// TaskAwareMOERouter for MI455X (gfx1250, wave32)
//   B=64, C=768, H=W=32, NUM_CLASSES=100, E=16, K=2, FAN_IN=868
// Dominant cost: 201MB avg-pool read (~8.6us @ 23.3TB/s). GEMM done with
// V_WMMA_F32_16X16X4_F32 (fp32 WMMA) per CDNA5 ISA 7.12.
// ---------------------------------------------------------------------------

#define Bsz 64
#define Cch 768
#define HW  1024
#define NCLS 100
#define NEXP 16
#define FANIN 868   // C + NUM_CLASSES

typedef __attribute__((ext_vector_type(2))) float v2f;
typedef __attribute__((ext_vector_type(4))) float v4f;
typedef __attribute__((ext_vector_type(8))) float v8f;

// ---------------------------------------------------------------------------
// Kernel 1: mean over H*W for each (b,c) row. One block per row, 256 threads,
// one float4 (b128) NT load per thread -> fully coalesced streaming read.
// Writes mean into fused[b*FANIN + c].
// ---------------------------------------------------------------------------
__global__ __launch_bounds__(256) void pool_mean_kernel(const float* __restrict__ hs,
                                                        float* __restrict__ fused) {
    const int row = blockIdx.x;                 // 0 .. B*C-1
    const int b = row / Cch;
    const int c = row - b * Cch;
    const v4f* p = (const v4f*)(hs + (size_t)row * HW);
    v4f v = __builtin_nontemporal_load(p + threadIdx.x);
    float s = v[0] + v[1] + v[2] + v[3];

    __shared__ float sm[256];
    sm[threadIdx.x] = s;
    __syncthreads();
    for (int off = 128; off > 0; off >>= 1) {
        if (threadIdx.x < off) sm[threadIdx.x] += sm[threadIdx.x + off];
        __syncthreads();
    }
    if (threadIdx.x == 0)
        fused[(size_t)b * FANIN + c] = sm[0] * (1.0f / (float)HW);
}

// ---------------------------------------------------------------------------
// Kernel 2: softmax(task_cls) per row -> fused[b*FANIN + 768 + j]
// One 128-thread block per batch row (100 active lanes).
// ---------------------------------------------------------------------------
__global__ __launch_bounds__(128) void task_softmax_kernel(const float* __restrict__ task,
                                                           float* __restrict__ fused) {
    const int b = blockIdx.x;
    const int t = threadIdx.x;
    __shared__ float s[128];

    float v = (t < NCLS) ? task[b * NCLS + t] : -INFINITY;
    s[t] = v;
    __syncthreads();
    for (int off = 64; off > 0; off >>= 1) {
        if (t < off) s[t] = fmaxf(s[t], s[t + off]);
        __syncthreads();
    }
    float m = s[0];
    __syncthreads();

    float e = (t < NCLS) ? __expf(v - m) : 0.0f;
    s[t] = e;
    __syncthreads();
    for (int off = 64; off > 0; off >>= 1) {
        if (t < off) s[t] += s[t + off];
        __syncthreads();
    }
    float sum = s[0];

    if (t < NCLS)
        fused[(size_t)b * FANIN + Cch + t] = e / sum;
}

// ---------------------------------------------------------------------------
// Kernel 3: gate GEMM  logits[64,16] = fused[64,868] x W^T[868,16] + bias
// using V_WMMA_F32_16X16X4_F32.  One workgroup, 4 waves; wave w owns M-tile
// rows [16w, 16w+16).  K-loop: 217 steps of K=4 (868 = 217*4 exactly).
//
// Operand layout (ISA 7.12.2, 32-bit A 16x4):
//   lane L: m = L&15, koff = 2*(L>>4); vgpr0 = A[m][k0+koff], vgpr1 = A[m][k0+koff+1]
// B (4x16) mirrors with n = L&15: vgpr0 = B[k0+koff][n] = W[n][k0+koff], etc.
// Both operands are consecutive-float pairs -> single b64 loads.
// D layout: lane group selects M in {r, 8+r}, col = lane&15.
// ---------------------------------------------------------------------------
__global__ __launch_bounds__(128) void router_gemm_kernel(const float* __restrict__ fused,
                                                          const float* __restrict__ Wm,
                                                          const float* __restrict__ bias,
                                                          float* __restrict__ out_logits) {
    const int lane = threadIdx.x & 31;
    const int wave = threadIdx.x >> 5;        // 0..3 -> M-tile
    const int n    = lane & 15;               // A: row-in-tile / B: expert column
    const int koff = (lane >> 4) << 1;        // 0 or 2

    const float* arow = fused + (size_t)(wave * 16 + n) * FANIN;
    const float* brow = Wm    + (size_t)n * FANIN;

    v8f acc = {};
    for (int k0 = 0; k0 < FANIN; k0 += 4) {
        float2 af = *(const float2*)(arow + k0 + koff);
        float2 bf = *(const float2*)(brow + k0 + koff);
        v2f a; a[0] = af.x; a[1] = af.y;
        v2f bb; bb[0] = bf.x; bb[1] = bf.y;
        acc = __builtin_amdgcn_wmma_f32_16x16x4_f32(
            /*neg_a=*/false, a, /*neg_b=*/false, bb,
            /*c_mod=*/(short)0, acc, /*reuse_a=*/false, /*reuse_b=*/false);
    }

    const float bv = bias[n];
    const int mbase = wave * 16 + ((lane >> 4) * 8);
#pragma unroll
    for (int r = 0; r < 8; ++r) {
        out_logits[(mbase + r) * NEXP + n] = acc[r] + bv;
    }
}

// ---------------------------------------------------------------------------
// Kernel 4: per-row softmax over E=16, top-2 (ties -> lowest index, matching
// jax.lax.top_k), renormalize, emit selected experts + one-hot mask [E,K,B].
// One thread per batch row; mask stores coalesced along B.
// ---------------------------------------------------------------------------
__global__ __launch_bounds__(64) void topk_finalize_kernel(const float* __restrict__ logits,
                                                           float* __restrict__ out) {
    const int bb = threadIdx.x;
    if (bb >= Bsz) return;

    float l[NEXP];
    float mx = -INFINITY;
#pragma unroll
    for (int e = 0; e < NEXP; ++e) {
        l[e] = logits[bb * NEXP + e];
        mx = fmaxf(mx, l[e]);
    }
    // top-1 (first occurrence wins on ties)
    int i1 = 0; float v1 = l[0];
#pragma unroll
    for (int e = 1; e < NEXP; ++e)
        if (l[e] > v1) { v1 = l[e]; i1 = e; }
    // top-2 excluding i1
    int i2 = -1; float v2 = -INFINITY;
#pragma unroll
    for (int e = 0; e < NEXP; ++e)
        if (e != i1 && l[e] > v2) { v2 = l[e]; i2 = e; }

    const float e1 = __expf(v1 - mx);
    const float e2 = __expf(v2 - mx);
    const float inv = 1.0f / (e1 + e2);

    // router_weights [B,2] at offset 1024
    out[1024 + bb * 2 + 0] = e1 * inv;
    out[1024 + bb * 2 + 1] = e2 * inv;
    // selected_experts [B,2] at offset 1152 (int values as float)
    out[1152 + bb * 2 + 0] = (float)i1;
    out[1152 + bb * 2 + 1] = (float)i2;
    // expert_mask [E,K,B] at offset 1280
#pragma unroll
    for (int e = 0; e < NEXP; ++e) {
        out[1280 + (e * 2 + 0) * Bsz + bb] = (i1 == e) ? 1.0f : 0.0f;
        out[1280 + (e * 2 + 1) * Bsz + bb] = (i2 == e) ? 1.0f : 0.0f;
    }
}

// ---------------------------------------------------------------------------
extern "C" void kernel_launch(void* const* d_in, const int* in_sizes, int n_in,
                              void* d_out, int out_size, void* d_ws, size_t ws_size,
                              hipStream_t stream) {
    const float* hs   = (const float*)d_in[0];   // [64,768,32,32]
    const float* task = (const float*)d_in[1];   // [64,100]
    const float* Wm   = (const float*)d_in[2];   // [16,868]
    const float* bias = (const float*)d_in[3];   // [16]
    float* out = (float*)d_out;                  // 3328 floats
    float* fused = (float*)d_ws;                 // [64,868] scratch (222 KB)

    // 1) streaming avg-pool: bandwidth-bound, ~8.6us floor @ 23.3 TB/s
    pool_mean_kernel<<<Bsz * Cch, 256, 0, stream>>>(hs, fused);
    // 2) task-logit softmax into fused[:, 768:868]
    task_softmax_kernel<<<Bsz, 128, 0, stream>>>(task, fused);
    // 3) gate GEMM via v_wmma_f32_16x16x4_f32 -> logits at out[0:1024]
    router_gemm_kernel<<<1, 128, 0, stream>>>(fused, Wm, bias, out);
    // 4) softmax/top-2/renorm/one-hot
    topk_finalize_kernel<<<1, 64, 0, stream>>>(out, out);
}